// HomoMGDN_3246995275926
// MI455X (gfx1250) — compile-verified
//
#include <hip/hip_runtime.h>
#include <stdint.h>

#define DFEAT 256            // feature dim (reference constant)
#define ALPHA_F 0.1f
#define BETA_F 0.9f
#define KITER 10

// ---------------------------------------------------------------------------
// CDNA5 async global->LDS copy helpers (ASYNCcnt-tracked, ISA 15.18 / ch.8)
// Each lane copies 32 bytes (2 x b128). INST_OFFSET applies to both the LDS
// destination and the global source address per the ISA pseudocode.
// TH_LOAD_HT: gathered node rows are reused ~degree times per iteration and
// the whole feature matrix fits in the 192MB L2 -> keep them high priority.
// ---------------------------------------------------------------------------
__device__ __forceinline__ void async_gather32(const float* gsrc, uint32_t ldsoff) {
  asm volatile(
      "global_load_async_to_lds_b128 %0, %1, off th:TH_LOAD_HT\n\t"
      "global_load_async_to_lds_b128 %0, %1, off offset:16 th:TH_LOAD_HT"
      :: "v"(ldsoff), "v"(gsrc)
      : "memory");
}
__device__ __forceinline__ void wait_async_le2() {
  asm volatile("s_wait_asynccnt 0x2" ::: "memory");
}
__device__ __forceinline__ void wait_async_le0() {
  asm volatile("s_wait_asynccnt 0x0" ::: "memory");
}

// ---------------------------------------------------------------------------
// Normalization kernels
// ---------------------------------------------------------------------------
__global__ void init_deg_kernel(float* __restrict__ deg, int N) {
  int i = blockIdx.x * blockDim.x + threadIdx.x;
  if (i < N) deg[i] = 1.0f;   // self-loop weight folded into degree init
}

__global__ void deg_accum_kernel(float* __restrict__ deg,
                                 const float* __restrict__ w,
                                 const int* __restrict__ row, int E) {
  int e = blockIdx.x * blockDim.x + threadIdx.x;
  if (e < E) atomicAdd(&deg[row[e]], w[e]);
}

__global__ void dinv_kernel(float* __restrict__ dinv,
                            const float* __restrict__ deg, int N) {
  int i = blockIdx.x * blockDim.x + threadIdx.x;
  if (i < N) {
    float d = deg[i];
    dinv[i] = (d > 0.0f) ? rsqrtf(d) : 0.0f;
  }
}

__global__ void wn_kernel(float* __restrict__ wn,
                          const float* __restrict__ w,
                          const int* __restrict__ row,
                          const int* __restrict__ col,
                          const float* __restrict__ dinv, int E) {
  int e = blockIdx.x * blockDim.x + threadIdx.x;
  if (e < E) wn[e] = dinv[row[e]] * w[e] * dinv[col[e]];
}

__global__ void zero_kernel(float4* __restrict__ p, long n4) {
  long i = (long)blockIdx.x * blockDim.x + threadIdx.x;
  long stride = (long)gridDim.x * blockDim.x;
  float4 z = make_float4(0.f, 0.f, 0.f, 0.f);
  for (; i < n4; i += stride) p[i] = z;
}

// ---------------------------------------------------------------------------
// Edge SpMM: one wave32 per edge; 32 lanes x 32B = one 1KB feature row.
// Double-buffered LDS staging via async copies: gather of edge e+1 (and its
// wn/row metadata) overlaps the atomic scatter of edge e. All control flow is
// wave-uniform so EXEC is all-ones when async ops issue.
// ---------------------------------------------------------------------------
__global__ __launch_bounds__(256) void spmm_kernel(
    const float* __restrict__ cur, float* __restrict__ acc,
    const float* __restrict__ wn, const int* __restrict__ row,
    const int* __restrict__ col, int E) {
  __shared__ __align__(16) float lds[8][2][DFEAT];   // 8 waves * 2 slots * 1KB
  const int lane = threadIdx.x & 31;
  const int wid  = threadIdx.x >> 5;
  const long nwaves = (long)gridDim.x * 8;
  long e = (long)blockIdx.x * 8 + wid;
  if (e >= E) return;                                 // wave-uniform

  const uint32_t chunk = (uint32_t)(lane * 32);
  uint32_t base[2];
  base[0] = (uint32_t)(uintptr_t)(&lds[wid][0][0]) + chunk;
  base[1] = (uint32_t)(uintptr_t)(&lds[wid][1][0]) + chunk;

  // prologue: issue gather of first edge's source row into slot 0, and load
  // its metadata so the steady-state loop never stalls on scalar data.
  {
    long c0 = col[e];
    async_gather32(cur + c0 * DFEAT + lane * 8, base[0]);
  }
  float wv = wn[e];
  long  r  = row[e];

  int slot = 0;
  for (; e < E; e += nwaves) {
    long  en   = e + nwaves;
    float wv_n = 0.0f;
    long  r_n  = 0;
    if (en < E) {
      long cn = col[en];
      async_gather32(cur + cn * DFEAT + lane * 8, base[slot ^ 1]);
      wv_n = wn[en];
      r_n  = row[en];
      wait_async_le2();   // async ops complete in-order: current slot is ready
    } else {
      wait_async_le0();
    }
    const float4* lp = (const float4*)(&lds[wid][slot][lane * 8]);
    float4 a = lp[0];
    float4 b = lp[1];
    float* dst = acc + r * DFEAT + lane * 8;
    atomicAdd(dst + 0, a.x * wv);
    atomicAdd(dst + 1, a.y * wv);
    atomicAdd(dst + 2, a.z * wv);
    atomicAdd(dst + 3, a.w * wv);
    atomicAdd(dst + 4, b.x * wv);
    atomicAdd(dst + 5, b.y * wv);
    atomicAdd(dst + 6, b.z * wv);
    atomicAdd(dst + 7, b.w * wv);
    wv = wv_n;
    r  = r_n;
    slot ^= 1;
  }
}

// ---------------------------------------------------------------------------
// Fused combine: acc = ((acc + dinv[r]^2 * cur) * BETA + ALPHA * x) * scale
// (self-loop contribution dinv^2*cur folded in here instead of the edge loop)
// ---------------------------------------------------------------------------
__global__ void combine_kernel(float* __restrict__ acc,
                               const float* __restrict__ cur,
                               const float* __restrict__ x,
                               const float* __restrict__ dinv,
                               long n4, float scale) {
  long i = (long)blockIdx.x * blockDim.x + threadIdx.x;
  long stride = (long)gridDim.x * blockDim.x;
  float4* a4 = (float4*)acc;
  const float4* c4 = (const float4*)cur;
  const float4* x4 = (const float4*)x;
  for (; i < n4; i += stride) {
    long node = i >> 6;                 // (i*4)/256
    float di = dinv[node];
    float sw = di * di;
    float4 a = a4[i], c = c4[i], h = x4[i];
    float4 o;
    o.x = ((a.x + sw * c.x) * BETA_F + ALPHA_F * h.x) * scale;
    o.y = ((a.y + sw * c.y) * BETA_F + ALPHA_F * h.y) * scale;
    o.z = ((a.z + sw * c.z) * BETA_F + ALPHA_F * h.z) * scale;
    o.w = ((a.w + sw * c.w) * BETA_F + ALPHA_F * h.w) * scale;
    a4[i] = o;
  }
}

// ---------------------------------------------------------------------------
// Host launcher
// ---------------------------------------------------------------------------
extern "C" void kernel_launch(void* const* d_in, const int* in_sizes, int n_in,
                              void* d_out, int out_size, void* d_ws, size_t ws_size,
                              hipStream_t stream) {
  const float* x  = (const float*)d_in[0];
  const float* ew = (const float*)d_in[1];
  const int*   ei = (const int*)d_in[2];

  const long ND = (long)in_sizes[0];      // N * 256
  const int  E  = in_sizes[1];
  const int  N  = (int)(ND / DFEAT);
  const int* row  = ei;
  const int* colv = ei + E;

  // workspace carve-up (floats): ping buffer A, wn[E], dinv[N], deg[N]
  float* wsf  = (float*)d_ws;
  float* A    = wsf;
  float* wn   = A + ND;
  float* dinv = wn + E;
  float* deg  = dinv + N;
  float* B    = (float*)d_out;            // iteration 9 (odd) accumulates here

  // ---- normalization ----
  init_deg_kernel<<<(N + 255) / 256, 256, 0, stream>>>(deg, N);
  deg_accum_kernel<<<(E + 255) / 256, 256, 0, stream>>>(deg, ew, row, E);
  dinv_kernel<<<(N + 255) / 256, 256, 0, stream>>>(dinv, deg, N);
  wn_kernel<<<(E + 255) / 256, 256, 0, stream>>>(wn, ew, row, colv, dinv, E);

  // GAMMA = beta^K + alpha * sum beta^i  (== 1.0 analytically; keep numeric)
  double bp = 1.0, s = 0.0;
  for (int i = 0; i < KITER; ++i) { s += bp; bp *= 0.9; }
  const double gamma = bp + 0.1 * s;
  const float invGamma = (float)(1.0 / gamma);

  const long n4 = ND / 4;
  for (int i = 0; i < KITER; ++i) {
    const float* cur = (i == 0) ? x : ((i & 1) ? A : B);
    float* acc = (i & 1) ? B : A;
    zero_kernel<<<2048, 256, 0, stream>>>((float4*)acc, n4);
    spmm_kernel<<<8192, 256, 0, stream>>>(cur, acc, wn, row, colv, E);
    combine_kernel<<<4096, 256, 0, stream>>>(acc, cur, x, dinv, n4,
                                             (i == KITER - 1) ? invGamma : 1.0f);
  }
}